// GAT_44822278701441
// MI455X (gfx1250) — compile-verified
//
#include <hip/hip_runtime.h>
#include <math.h>

typedef __attribute__((ext_vector_type(2))) float v2f;
typedef __attribute__((ext_vector_type(8))) float v8f;

#define DEVFN __device__ __forceinline__

constexpr int N_   = 50000;
constexpr int E_   = 800000;
constexpr int EP_  = E_ + N_;     // edges incl. self-loops
constexpr int INC  = 256;
constexpr int HID  = 256;
constexpr int OUTC = 64;
constexpr float SLOPE = 0.2f;

// ---------- order-preserving float<->uint for atomic max ----------
DEVFN unsigned f2ord(float f) {
  unsigned u = __float_as_uint(f);
  return (u & 0x80000000u) ? ~u : (u | 0x80000000u);
}
DEVFN float ord2f(unsigned u) {
  return __uint_as_float((u & 0x80000000u) ? (u & 0x7fffffffu) : ~u);
}

DEVFN void edge_sd(const int* __restrict__ ei, int idx, int& s, int& d) {
  if (idx < E_) { s = ei[idx]; d = ei[E_ + idx]; }
  else          { s = idx - E_; d = s; }            // self loop
}

// ---------- zero fill ----------
__global__ void fill0_kernel(unsigned* __restrict__ p, long n) {
  long i = (long)blockIdx.x * blockDim.x + threadIdx.x;
  if (i < n) p[i] = 0u;
}

// ---------- fp32 WMMA GEMM: C[M,N] = A[M,K] x B[K,N], row major ----------
// 256 threads = 8 waves; block covers 16 rows x 128 cols; wave -> 16x16 tile.
// A tile staged in LDS (stride 260 floats: conflict-free, 8-byte aligned rows).
__global__ void gemm_wmma(const float* __restrict__ A, const float* __restrict__ B,
                          float* __restrict__ C, int K, int N) {
  __shared__ float As[16 * 260];
  const int row0 = blockIdx.x * 16;
  for (int t = threadIdx.x; t < 16 * K; t += blockDim.x) {
    int r = t / K, k = t - r * K;
    As[r * 260 + k] = A[(long)(row0 + r) * K + k];
  }
  __syncthreads();

  const int wave = threadIdx.x >> 5;
  const int lane = threadIdx.x & 31;
  const int half = lane >> 4;      // K sub-pair selector (ISA 16x4 f32 A layout)
  const int l16  = lane & 15;
  const int col0 = blockIdx.y * 128 + wave * 16;
  if (col0 >= N) return;           // wave-uniform: EXEC stays all-ones for WMMA

  v8f acc = {};
  const float* Arow = &As[l16 * 260];
  for (int k = 0; k < K; k += 4) {
    const int ka = k + half * 2;
    v2f a, b;
    a.x = Arow[ka];
    a.y = Arow[ka + 1];
    b.x = B[(long)ka       * N + col0 + l16];
    b.y = B[(long)(ka + 1) * N + col0 + l16];
    acc = __builtin_amdgcn_wmma_f32_16x16x4_f32(
        /*neg_a=*/false, a, /*neg_b=*/false, b,
        /*c_mod=*/(short)0, acc, /*reuse_a=*/false, /*reuse_b=*/false);
  }
#pragma unroll
  for (int r = 0; r < 8; ++r)
    C[(long)(row0 + half * 8 + r) * N + col0 + l16] = acc[r];
}

// ---------- per-(node,head) attention coefficients ----------
__global__ void attn_coef(const float* __restrict__ h, const float* __restrict__ asrc,
                          const float* __restrict__ adst, float* __restrict__ als,
                          float* __restrict__ ald, int H, int C) {
  int t = blockIdx.x * blockDim.x + threadIdx.x;
  if (t >= N_ * H) return;
  int n = t / H, hd = t - n * H;
  const float* hp = h + (long)n * H * C + (long)hd * C;
  const float* ap = asrc + hd * C;
  const float* bp = adst + hd * C;
  float s = 0.f, d = 0.f;
  for (int c = 0; c < C; ++c) { float v = hp[c]; s = fmaf(v, ap[c], s); d = fmaf(v, bp[c], d); }
  als[t] = s; ald[t] = d;
}

// ---------- edge pass 1: leaky-relu score + segment max ----------
__global__ void edge_max(const int* __restrict__ ei, const float* __restrict__ als,
                         const float* __restrict__ ald, float* __restrict__ ebuf,
                         unsigned* __restrict__ mord, int H) {
  long t = (long)blockIdx.x * blockDim.x + threadIdx.x;
  if (t >= (long)EP_ * H) return;
  int idx = (int)(t / H), hd = (int)(t - (long)idx * H);
  int s, d; edge_sd(ei, idx, s, d);
  float e = als[s * H + hd] + ald[d * H + hd];
  e = e > 0.f ? e : SLOPE * e;
  ebuf[t] = e;
  atomicMax(&mord[d * H + hd], f2ord(e));
}

// ---------- edge pass 2: exp + segment sum ----------
__global__ void edge_expsum(const int* __restrict__ ei, float* __restrict__ ebuf,
                            const unsigned* __restrict__ mord, float* __restrict__ ssum,
                            int H) {
  long t = (long)blockIdx.x * blockDim.x + threadIdx.x;
  if (t >= (long)EP_ * H) return;
  int idx = (int)(t / H), hd = (int)(t - (long)idx * H);
  int s, d; edge_sd(ei, idx, s, d);
  (void)s;
  float ex = __expf(ebuf[t] - ord2f(mord[d * H + hd]));
  ebuf[t] = ex;
  atomicAdd(&ssum[d * H + hd], ex);
}

// ---------- edge pass 3: weighted scatter-add of features ----------
// thread per (edge, feature); F = H*C = 1<<FSH, C = 1<<CSH.
__global__ void edge_agg(const int* __restrict__ ei, const float* __restrict__ h,
                         const float* __restrict__ ebuf, const float* __restrict__ ssum,
                         float* __restrict__ agg, int H, int CSH, int FSH) {
  long t = (long)blockIdx.x * blockDim.x + threadIdx.x;
  if (t >= ((long)EP_ << FSH)) return;
  int idx = (int)(t >> FSH);
  int f   = (int)(t & ((1 << FSH) - 1));
  int hd  = f >> CSH;
  int s, d; edge_sd(ei, idx, s, d);
  float w = ebuf[(long)idx * H + hd] / (ssum[d * H + hd] + 1e-16f);
  atomicAdd(&agg[((long)d << FSH) + f], h[((long)s << FSH) + f] * w);
}

// ---------- bias + optional ELU (mode 1) ----------
__global__ void bias_act(const float* __restrict__ agg, const float* __restrict__ b,
                         float* __restrict__ out, int F, int mode) {
  long t = (long)blockIdx.x * blockDim.x + threadIdx.x;
  if (t >= (long)N_ * F) return;
  int f = (int)(t % F);
  float v = agg[t] + b[f];
  if (mode) v = v > 0.f ? v : (__expf(v) - 1.f);
  out[t] = v;
}

static inline unsigned cdiv(long a, long b) { return (unsigned)((a + b - 1) / b); }

extern "C" void kernel_launch(void* const* d_in, const int* in_sizes, int n_in,
                              void* d_out, int out_size, void* d_ws, size_t ws_size,
                              hipStream_t stream) {
  (void)in_sizes; (void)n_in; (void)out_size; (void)ws_size;
  const float* x    = (const float*)d_in[0];
  const int*   ei   = (const int*)d_in[1];
  const float* W1   = (const float*)d_in[2];
  const float* at_s1= (const float*)d_in[3];
  const float* at_d1= (const float*)d_in[4];
  const float* b1   = (const float*)d_in[5];
  const float* W2   = (const float*)d_in[6];
  const float* at_s2= (const float*)d_in[7];
  const float* at_d2= (const float*)d_in[8];
  const float* b2   = (const float*)d_in[9];
  float* out = (float*)d_out;

  // workspace layout
  char* p = (char*)d_ws;
  float*    h1   = (float*)p;    p += sizeof(float) * (size_t)N_ * HID;   // 12.8M f
  float*    agg1 = (float*)p;    p += sizeof(float) * (size_t)N_ * HID;   // -> h1act
  float*    e1   = (float*)p;    p += sizeof(float) * (size_t)EP_ * 8;
  float*    als1 = (float*)p;    p += sizeof(float) * (size_t)N_ * 8;
  float*    ald1 = (float*)p;    p += sizeof(float) * (size_t)N_ * 8;
  unsigned* m1   = (unsigned*)p; p += sizeof(unsigned) * (size_t)N_ * 8;
  float*    s1   = (float*)p;    p += sizeof(float) * (size_t)N_ * 8;
  // layer-2 reuses layer-1 regions that are dead by then
  float*    h2   = h1;                   // gemm2 reads agg1 (h1act), writes h1 region
  float*    e2   = e1;
  float*    als2 = als1;
  float*    ald2 = ald1;
  unsigned* m2   = m1;
  float*    s2   = s1;

  const int B = 256;

  // ----- zero accumulators (every call: deterministic) -----
  fill0_kernel<<<cdiv((long)N_ * 8 * 2, B), B, 0, stream>>>(m1, (long)N_ * 8);        // m1
  fill0_kernel<<<cdiv((long)N_ * 8, B), B, 0, stream>>>((unsigned*)s1, (long)N_ * 8); // s1
  fill0_kernel<<<cdiv((long)N_ * HID, B), B, 0, stream>>>((unsigned*)agg1, (long)N_ * HID);
  fill0_kernel<<<cdiv((long)N_ * OUTC, B), B, 0, stream>>>((unsigned*)out, (long)N_ * OUTC);

  // ----- layer 1 -----
  gemm_wmma<<<dim3(N_ / 16, HID / 128), B, 0, stream>>>(x, W1, h1, INC, HID);
  attn_coef<<<cdiv((long)N_ * 8, B), B, 0, stream>>>(h1, at_s1, at_d1, als1, ald1, 8, 32);
  edge_max   <<<cdiv((long)EP_ * 8, B), B, 0, stream>>>(ei, als1, ald1, e1, m1, 8);
  edge_expsum<<<cdiv((long)EP_ * 8, B), B, 0, stream>>>(ei, e1, m1, s1, 8);
  edge_agg   <<<cdiv((long)EP_ * 256, B), B, 0, stream>>>(ei, h1, e1, s1, agg1, 8, 5, 8);
  bias_act   <<<cdiv((long)N_ * HID, B), B, 0, stream>>>(agg1, b1, agg1, HID, 1); // ELU in place

  // ----- layer 2 -----
  gemm_wmma<<<dim3(N_ / 16, 1), B, 0, stream>>>(agg1, W2, h2, HID, OUTC);
  fill0_kernel<<<cdiv((long)N_, B), B, 0, stream>>>(m2, (long)N_);
  fill0_kernel<<<cdiv((long)N_, B), B, 0, stream>>>((unsigned*)s2, (long)N_);
  attn_coef<<<cdiv((long)N_, B), B, 0, stream>>>(h2, at_s2, at_d2, als2, ald2, 1, 64);
  edge_max   <<<cdiv((long)EP_, B), B, 0, stream>>>(ei, als2, ald2, e2, m2, 1);
  edge_expsum<<<cdiv((long)EP_, B), B, 0, stream>>>(ei, e2, m2, s2, 1);
  edge_agg   <<<cdiv((long)EP_ * 64, B), B, 0, stream>>>(ei, h2, e2, s2, out, 1, 6, 6);
  bias_act   <<<cdiv((long)N_ * OUTC, B), B, 0, stream>>>(out, b2, out, OUTC, 0); // + b2 in place
}